// LaplacianNorm_26482768347668
// MI455X (gfx1250) — compile-verified
//
#include <hip/hip_runtime.h>
#include <stdint.h>

typedef __attribute__((ext_vector_type(2))) float v2f;
typedef __attribute__((ext_vector_type(8))) float v8f;

#define LN_EPS 1e-6f
#define D_FEAT 128
#define D4     32              // float4s per row
#define TILE_ROWS 16
#define ROW_PAD 132            // padded LDS row stride (floats): conflict-free A reads
#define WAVES_PER_BLOCK 4

// ---------------- K0: zero accumulators ----------------
__global__ void ln_zero_kernel(float* __restrict__ p, int total) {
    int i = blockIdx.x * blockDim.x + threadIdx.x;
    if (i < total) p[i] = 0.0f;
}

// ---------------- K1: degree accumulation ----------------
__global__ void ln_deg_kernel(const float* __restrict__ ev, const int* __restrict__ row,
                              float* __restrict__ deg, int e) {
    int i = blockIdx.x * blockDim.x + threadIdx.x;
    if (i < e) atomicAdd(&deg[row[i]], ev[i]);
}

// ---------------- K2: per-node row stats via async-LDS staging + WMMA (B = ones) ----
// xmean[n] = (1/128) sum_d x[n,d] ; sqmean[n] = (1/128) sum_d x[n,d]^2
// disqrt[n] = rsqrt(deg[n] + eps)
__global__ void ln_rowstats_kernel(const float* __restrict__ x,
                                   const float* __restrict__ deg,
                                   float* __restrict__ xmean,
                                   float* __restrict__ sqmean,
                                   float* __restrict__ disqrt,
                                   int n) {
    __shared__ float lds[WAVES_PER_BLOCK * TILE_ROWS * ROW_PAD];
    const int lane = threadIdx.x & 31;
    const int wave = threadIdx.x >> 5;
    const int base = (blockIdx.x * WAVES_PER_BLOCK + wave) * TILE_ROWS;
    float* tileLds = &lds[wave * TILE_ROWS * ROW_PAD];

    // Stage 16 rows x 128 floats straight into LDS via the async engine
    // (GLOBAL_LOAD_ASYNC_TO_LDS_B128, tracked by ASYNCcnt — no VGPR round-trip).
    #pragma unroll
    for (int r = 0; r < TILE_ROWS; ++r) {
        int node = base + r;
        if (node < n) {
            const float* src = x + (size_t)node * D_FEAT + lane * 4;
            // generic LDS pointer: addr[31:0] is the workgroup-relative LDS offset
            unsigned dstOff = (unsigned)(uintptr_t)(tileLds + r * ROW_PAD + lane * 4);
            asm volatile("global_load_async_to_lds_b128 %0, %1, off"
                         :: "v"(dstOff), "v"(src)
                         : "memory");
        } else {
            // zero-fill OOB tail rows so the A-matrix stays clean
            float4 z = make_float4(0.f, 0.f, 0.f, 0.f);
            ((float4*)(tileLds + r * ROW_PAD))[lane] = z;
        }
    }
    asm volatile("s_wait_asynccnt 0x0" ::: "memory");
    __syncthreads();

    // A-matrix 16x4 f32 layout: lane L supplies A[L&15, 2*(L>>4) + {0,1}] in VGPRs {0,1}
    const int m = lane & 15;
    const int half = lane >> 4;
    const float* aRow = tileLds + m * ROW_PAD + 2 * half;

    v8f csum = {};
    v8f csq  = {};
    v2f ones2; ones2[0] = 1.0f; ones2[1] = 1.0f;   // B = ones(4x16): layout-invariant

    #pragma unroll
    for (int k0 = 0; k0 < D_FEAT; k0 += 4) {
        v2f a = *(const v2f*)(aRow + k0);           // ds_load_b64, conflict-free
        v2f asq = a * a;
        // D[m,*] = sum_k A[m,k] accumulated into C (full f32 precision)
        csum = __builtin_amdgcn_wmma_f32_16x16x4_f32(false, a,   false, ones2,
                                                     (short)0, csum, false, false);
        csq  = __builtin_amdgcn_wmma_f32_16x16x4_f32(false, asq, false, ones2,
                                                     (short)0, csq,  false, false);
    }

    // C/D layout: VGPR j @ lane L = D[j + 8*(L>=16), L&15]; all N columns identical.
    const int j = lane & 7;
    float ssum = csum[0], ssq = csq[0];
    #pragma unroll
    for (int t = 1; t < 8; ++t) {
        ssum = (j == t) ? csum[t] : ssum;
        ssq  = (j == t) ? csq[t]  : ssq;
    }
    // lanes 0-7 hold rows 0-7, lanes 16-23 hold rows 8-15
    int node = base + j + 8 * half;
    if (((lane & 8) == 0) && node < n) {
        xmean[node]  = ssum * (1.0f / (float)D_FEAT);
        sqmean[node] = ssq  * (1.0f / (float)D_FEAT);
        disqrt[node] = rsqrtf(deg[node] + LN_EPS);
    }
}

// ---------------- K3: edge pass 1 — normalized weights + scalar mean ----------------
__global__ void ln_edge_mean_kernel(const float* __restrict__ ev,
                                    const int* __restrict__ row, const int* __restrict__ col,
                                    const float* __restrict__ disqrt,
                                    const float* __restrict__ xmean,
                                    float* __restrict__ w, float* __restrict__ meanAcc, int e) {
    int i = blockIdx.x * blockDim.x + threadIdx.x;
    if (i >= e) return;
    int r = row[i], c = col[i];
    float we = disqrt[r] * ev[i] * disqrt[c];
    w[i] = we;
    atomicAdd(&meanAcc[r], we * xmean[c]);
}

// ---------------- K4: per-node centered second moment ----------------
__global__ void ln_s2_kernel(const float* __restrict__ sqmean, const float* __restrict__ xmean,
                             const float* __restrict__ meanAcc, float* __restrict__ s2, int n) {
    int i = blockIdx.x * blockDim.x + threadIdx.x;
    if (i >= n) return;
    float mu = meanAcc[i];
    s2[i] = sqmean[i] - 2.0f * mu * xmean[i] + mu * mu;
}

// ---------------- K5: edge pass 2 — scalar variance ----------------
__global__ void ln_edge_var_kernel(const int* __restrict__ row, const int* __restrict__ col,
                                   const float* __restrict__ w, const float* __restrict__ s2,
                                   float* __restrict__ varAcc, int e) {
    int i = blockIdx.x * blockDim.x + threadIdx.x;
    if (i < e) atomicAdd(&varAcc[row[i]], w[i] * s2[col[i]]);
}

// ---------------- K6: fused normalize (float4 I/O) ----------------
__global__ void ln_finalize_kernel(const float* __restrict__ x,
                                   const float* __restrict__ meanAcc,
                                   const float* __restrict__ varAcc,
                                   const float* __restrict__ scale,
                                   const float* __restrict__ bias,
                                   float* __restrict__ out, int n) {
    int t = blockIdx.x * blockDim.x + threadIdx.x;
    int node = t >> 5;           // 32 float4s per row
    int d4 = t & 31;
    if (node >= n) return;
    float mu = meanAcc[node];
    float rstd = rsqrtf(varAcc[node] + LN_EPS);
    float4 xv = ((const float4*)(x + (size_t)node * D_FEAT))[d4];
    float4 sv = ((const float4*)scale)[d4];
    float4 bv = ((const float4*)bias)[d4];
    float4 o;
    o.x = (xv.x - mu) * rstd * sv.x + bv.x;
    o.y = (xv.y - mu) * rstd * sv.y + bv.y;
    o.z = (xv.z - mu) * rstd * sv.z + bv.z;
    o.w = (xv.w - mu) * rstd * sv.w + bv.w;
    ((float4*)(out + (size_t)node * D_FEAT))[d4] = o;
}

extern "C" void kernel_launch(void* const* d_in, const int* in_sizes, int n_in,
                              void* d_out, int out_size, void* d_ws, size_t ws_size,
                              hipStream_t stream) {
    const float* x     = (const float*)d_in[0];
    const float* ev    = (const float*)d_in[1];
    const float* scale = (const float*)d_in[2];
    const float* bias  = (const float*)d_in[3];
    const int*   row   = (const int*)d_in[4];
    const int*   col   = (const int*)d_in[5];
    const int n = in_sizes[0] / D_FEAT;
    const int e = in_sizes[1];
    float* out = (float*)d_out;

    // Workspace layout (floats): deg | meanAcc | varAcc | disqrt | xmean | sqmean | s2 | w
    float* ws      = (float*)d_ws;
    float* deg     = ws;
    float* meanAcc = ws + (size_t)n;
    float* varAcc  = ws + 2 * (size_t)n;
    float* disqrt  = ws + 3 * (size_t)n;
    float* xmean   = ws + 4 * (size_t)n;
    float* sqmean  = ws + 5 * (size_t)n;
    float* s2      = ws + 6 * (size_t)n;
    float* w       = ws + 7 * (size_t)n;

    const int B = 256;
    // K0: zero deg/meanAcc/varAcc (contiguous 3N floats)
    ln_zero_kernel<<<(3 * n + B - 1) / B, B, 0, stream>>>(deg, 3 * n);
    // K1: degree
    ln_deg_kernel<<<(e + B - 1) / B, B, 0, stream>>>(ev, row, deg, e);
    // K2: WMMA row stats + rsqrt(deg), async-LDS staging
    {
        int nodesPerBlock = WAVES_PER_BLOCK * TILE_ROWS;   // 64
        int grid = (n + nodesPerBlock - 1) / nodesPerBlock;
        ln_rowstats_kernel<<<grid, WAVES_PER_BLOCK * 32, 0, stream>>>(
            x, deg, xmean, sqmean, disqrt, n);
    }
    // K3: edge weights + mean
    ln_edge_mean_kernel<<<(e + B - 1) / B, B, 0, stream>>>(ev, row, col, disqrt, xmean, w, meanAcc, e);
    // K4: s2
    ln_s2_kernel<<<(n + B - 1) / B, B, 0, stream>>>(sqmean, xmean, meanAcc, s2, n);
    // K5: variance
    ln_edge_var_kernel<<<(e + B - 1) / B, B, 0, stream>>>(row, col, w, s2, varAcc, e);
    // K6: normalize
    {
        long long threads = (long long)n * D4;
        ln_finalize_kernel<<<(unsigned)((threads + B - 1) / B), B, 0, stream>>>(
            x, meanAcc, varAcc, scale, bias, out, n);
    }
}